// Renderer_2946347565328
// MI455X (gfx1250) — compile-verified
//
#include <hip/hip_runtime.h>
#include <hip/hip_bf16.h>
#include <stdint.h>

// Problem constants (from reference)
#define NCC    52
#define NRR    4
#define NTT    16
#define NSAMP  128
#define NRAYS  64
#define Q      64                 // NR*NT, contiguous fastest dims
#define X_S    (3 * NCC * Q)      // 9984 floats: x stride along sample axis
#define X_RAY  (NSAMP * X_S)
#define Z_S    (2 * NCC * Q)      // 6656 floats: z stride along sample axis
#define Z_RAY  (NSAMP * Z_S)
#define SECOFF (NCC * Q)          // 3328 floats between x sections (re / im / alpha)

#define TILE   8
#define NTILES (NSAMP / TILE)     // 16
// LDS stage layout (floats): x rows (sec*8+j)*64 -> 1536 floats; z rows 1536 + j*64 (j=0..8)
#define STAGE_F 2112
#define LDS_F   (2 * STAGE_F)

#define C_LIGHT 299792458.0f
#define PI_F    3.14159265358979323846f

#if defined(__AMDGCN__) && defined(__gfx1250__) && \
    __has_builtin(__builtin_amdgcn_global_load_async_to_lds_b128) && \
    __has_builtin(__builtin_amdgcn_s_wait_asynccnt)
#define USE_ASYNC 1
#else
#define USE_ASYNC 0
#endif

typedef __attribute__((ext_vector_type(4))) int v4i;
#if USE_ASYNC
typedef __attribute__((address_space(1))) v4i* g_v4i_ptr;
typedef __attribute__((address_space(3))) v4i* l_v4i_ptr;
#endif

__device__ __forceinline__ void async_row_b128(const float* g, float* lds_dst) {
#if USE_ASYNC
  // gfx1250: GLOBAL_LOAD_ASYNC_TO_LDS_B128, tracked with ASYNCcnt.
  __builtin_amdgcn_global_load_async_to_lds_b128(
      (g_v4i_ptr)g, (l_v4i_ptr)lds_dst, /*offset=*/0, /*cpol=*/0);
#else
  float4 v = *(const float4*)g;
  *(float4*)lds_dst = v;
#endif
}

template <int N>
__device__ __forceinline__ void wait_async() {
#if USE_ASYNC
  __builtin_amdgcn_s_wait_asynccnt(N);
#endif
}

__global__ __launch_bounds__(64) void renderer_scan_kernel(
    const float* __restrict__ x, const float* __restrict__ zv,
    const float* __restrict__ fc_vec, float* __restrict__ out) {
  __shared__ float smem[LDS_F];

  const int tid = threadIdx.x;       // 0..63, == flat (nr,nt) index q
  const int c   = blockIdx.x;        // channel 0..51
  const int ray = blockIdx.y;        // ray 0..63

  const size_t xbase = (size_t)ray * X_RAY + (size_t)c * Q;
  const size_t zbase = (size_t)ray * Z_RAY + (size_t)c * Q;

  // ---- producer: per-lane row assignment (16 lanes x 4 rows per async group)
  const int l = tid & 15;            // 16B chunk within a 256B row
  const int r = tid >> 4;            // row-in-group 0..3

  const float* xg[6];                // 24 x rows: idx = j*3 + sec
  uint32_t     xl[6];
#pragma unroll
  for (int g = 0; g < 6; ++g) {
    int idx = 4 * g + r;
    int j   = idx / 3;
    int sec = idx - 3 * j;
    xg[g] = x + xbase + (size_t)j * X_S + (size_t)sec * SECOFF + 4 * l;
    xl[g] = (uint32_t)((sec * 8 + j) * 64 + 4 * l);
  }
  const float* zg[2];                // z rows j=0..7
  uint32_t     zl[2];
#pragma unroll
  for (int g = 0; g < 2; ++g) {
    int j = 4 * g + r;
    zg[g] = zv + zbase + (size_t)j * Z_S + 4 * l;
    zl[g] = (uint32_t)(1536 + j * 64 + 4 * l);
  }
  // z overlap row (sample 8t+8, clamped to 127): all 64 lanes, 4x duplicated -> uniform ASYNCcnt
  const float*   z8  = zv + zbase + 4 * l;
  const uint32_t z8l = (uint32_t)(1536 + 8 * 64 + 4 * l);

  auto issue_tile = [&](int t, int stage) {
    const size_t xoff = (size_t)t * (TILE * X_S);
    const size_t zoff = (size_t)t * (TILE * Z_S);
    float* sb = smem + stage * STAGE_F;
#pragma unroll
    for (int g = 0; g < 6; ++g) async_row_b128(xg[g] + xoff, sb + xl[g]);
#pragma unroll
    for (int g = 0; g < 2; ++g) async_row_b128(zg[g] + zoff, sb + zl[g]);
    int s8 = t * TILE + TILE;
    if (s8 > NSAMP - 1) s8 = NSAMP - 1;  // duplicate row; never read for last tile
    async_row_b128(z8 + (size_t)s8 * Z_S, sb + z8l);
  };

  // ---- per-block scalars
  const float fc   = fc_vec[NCC + c];                       // fc_down
  const float wk   = 2.0f * PI_F * fc * 1.0e9f / C_LIGHT;   // wavenumber
  const float ampk = C_LIGHT / (fc * 1.0e9f * 4.0f * PI_F); // amp_decay = ampk / z

  float Pre = 1.0f, Pim = 0.0f;       // running exclusive product
  float accre = 0.0f, accim = 0.0f;
  const int q = tid;

  issue_tile(0, 0);
  for (int t = 0; t < NTILES; ++t) {
    __syncthreads();  // all waves done reading stage (t+1)&1 (tile t-1): safe to overwrite
    if (t + 1 < NTILES) {
      issue_tile(t + 1, (t + 1) & 1);
      wait_async<9>();  // in-order completion: <=9 outstanding => tile t resident
    } else {
      wait_async<0>();
    }
    __syncthreads();  // tile t visible to both waves

    const float* st = smem + (t & 1) * STAGE_F;
#pragma unroll
    for (int j = 0; j < TILE; ++j) {
      const int s   = t * TILE + j;
      const float cre = st[(0 * 8 + j) * 64 + q];
      const float cim = st[(1 * 8 + j) * 64 + q];
      const float ar  = st[(2 * 8 + j) * 64 + q];
      const float zc  = st[1536 + j * 64 + q];
      const float zn  = st[1536 + (j + 1) * 64 + q];

      const float alpha = 1.0f - __expf(-ar);
      const float dist  = (s == NSAMP - 1) ? 1e-10f : (zn - zc);
      float sn, cs;
      __sincosf(wk * dist, &sn, &cs);
      const float om  = 1.0f - alpha;
      const float Tre = __fmaf_rn(om, cs, 1e-10f);   // (1-a)*cos(kd) + 1e-10
      const float Tim = -om * sn;                    // -(1-a)*sin(kd)

      if (s != 0) {                                  // cumprod_exclusive[0] == 0
        const float aa  = alpha * (ampk / zc);
        const float wre = cre * Pre - cim * Pim;     // ch * P
        const float wim = cre * Pim + cim * Pre;
        accre = __fmaf_rn(aa, wre, accre);
        accim = __fmaf_rn(aa, wim, accim);
      }
      const float nre = Pre * Tre - Pim * Tim;       // P *= T
      const float nim = Pre * Tim + Pim * Tre;
      Pre = nre;
      Pim = nim;
    }
  }

  // out[ray][c][q], complex64 interleaved (re, im)
  const size_t o = (((size_t)ray * NCC) + c) * Q + q;
  out[2 * o]     = accre;
  out[2 * o + 1] = accim;
}

extern "C" void kernel_launch(void* const* d_in, const int* in_sizes, int n_in,
                              void* d_out, int out_size, void* d_ws, size_t ws_size,
                              hipStream_t stream) {
  (void)in_sizes; (void)n_in; (void)out_size; (void)d_ws; (void)ws_size;
  const float* x      = (const float*)d_in[0];
  const float* z_vals = (const float*)d_in[1];
  const float* fc_vec = (const float*)d_in[2];
  float*       out    = (float*)d_out;

  dim3 grid(NCC, NRAYS);   // 52 x 64 blocks, one (channel, ray) each
  dim3 block(Q);           // 64 threads = 2 wave32
  renderer_scan_kernel<<<grid, block, 0, stream>>>(x, z_vals, fc_vec, out);
}